// GraphAttentionLayer_36850819400289
// MI455X (gfx1250) — compile-verified
//
#include <hip/hip_runtime.h>
#include <math.h>

#define NV 8192
#define F 256
#define ALPHA 0.2f
#define JT 32
#define PT_STRIDE 49
#define WH_STRIDE 260

typedef __attribute__((ext_vector_type(2))) float v2f;
typedef __attribute__((ext_vector_type(8))) float v8f;

// ---------------------------------------------------------------------------
// Kernel A: Wh = h @ W   (f32 WMMA 16x16x4)
// Each workgroup: 16 rows x 256 cols. 4 waves, each wave owns 64 cols (4 tiles).
// ---------------------------------------------------------------------------
__global__ __launch_bounds__(128)
void gat_gemm_hW(const float* __restrict__ h, const float* __restrict__ W,
                 float* __restrict__ Wh) {
  const int t     = threadIdx.x;
  const int wv    = t >> 5;        // wave 0..3 -> column quarter
  const int lane  = t & 31;
  const int lhalf = lane >> 4;     // 0: K={0,1}, 1: K={2,3} (A/B frag halves)
  const int lm    = lane & 15;
  const int rb    = blockIdx.x * 16;
  const int nb    = wv * 64;

  v8f acc[4];
  #pragma unroll
  for (int i = 0; i < 4; ++i)
    acc[i] = (v8f){0.f,0.f,0.f,0.f,0.f,0.f,0.f,0.f};

  for (int k = 0; k < F; k += 4) {
    const int kr = k + lhalf * 2;
    v2f afrag;                      // A 16x4: lane holds row rb+lm, two K values
    afrag.x = h[(size_t)(rb + lm) * F + kr];
    afrag.y = h[(size_t)(rb + lm) * F + kr + 1];
    #pragma unroll
    for (int nt = 0; nt < 4; ++nt) {
      const int col = nb + nt * 16 + lm;
      v2f bfrag;                    // B 4x16: lane holds col, two K rows
      bfrag.x = W[(size_t)kr * F + col];
      bfrag.y = W[(size_t)(kr + 1) * F + col];
      acc[nt] = __builtin_amdgcn_wmma_f32_16x16x4_f32(
          false, afrag, false, bfrag, (short)0, acc[nt], false, false);
    }
  }

  #pragma unroll
  for (int nt = 0; nt < 4; ++nt) {
    #pragma unroll
    for (int v = 0; v < 8; ++v) {
      const int M = v + lhalf * 8;  // C/D layout: lanes>=16 hold M+8
      Wh[(size_t)(rb + M) * F + nb + nt * 16 + lm] = acc[nt][v];
    }
  }
}

// ---------------------------------------------------------------------------
// Kernel C: s1[i] = Wh[i,:] . a[:256] ; s2[i] = Wh[i,:] . a[256:512]
// One wave32 per row; shfl_xor tree reduction.
// ---------------------------------------------------------------------------
__global__ __launch_bounds__(256)
void gat_scores(const float* __restrict__ Wh, const float* __restrict__ a,
                float* __restrict__ s1, float* __restrict__ s2) {
  const int lane = threadIdx.x & 31;
  const int row  = blockIdx.x * 8 + (threadIdx.x >> 5);
  float p1 = 0.f, p2 = 0.f;
  for (int k = lane; k < F; k += 32) {
    const float w = Wh[(size_t)row * F + k];
    p1 += w * a[k];
    p2 += w * a[F + k];
  }
  #pragma unroll
  for (int off = 16; off > 0; off >>= 1) {
    p1 += __shfl_xor(p1, off, 32);
    p2 += __shfl_xor(p2, off, 32);
  }
  if (lane == 0) { s1[row] = p1; s2[row] = p2; }
}

// ---------------------------------------------------------------------------
// Kernel D: smax = max(s2)   (single workgroup)
// ---------------------------------------------------------------------------
__global__ __launch_bounds__(256)
void gat_s2max(const float* __restrict__ s2, float* __restrict__ smax) {
  __shared__ float red[256];
  float m = -3.0e38f;
  for (int i = threadIdx.x; i < NV; i += 256) m = fmaxf(m, s2[i]);
  red[threadIdx.x] = m;
  __syncthreads();
  for (int s = 128; s > 0; s >>= 1) {
    if (threadIdx.x < s) red[threadIdx.x] = fmaxf(red[threadIdx.x], red[threadIdx.x + s]);
    __syncthreads();
  }
  if (threadIdx.x == 0) smax[0] = red[0];
}

// ---------------------------------------------------------------------------
// Kernel B: fused masked softmax + att@Wh + ELU, flash style, no rescaling.
// Workgroup = 256 threads (8 waves) handles 32 output rows x 256 cols.
// Double-buffered Wh tiles staged via GLOBAL_LOAD_ASYNC_TO_LDS_B128 (ASYNCcnt)
// so the DMA for tile i+1 overlaps scores+WMMA of tile i.
// ---------------------------------------------------------------------------
__global__ __launch_bounds__(256)
void gat_attn(const float* __restrict__ Wh, const int* __restrict__ adj,
              const float* __restrict__ s1, const float* __restrict__ s2,
              const float* __restrict__ smax, float* __restrict__ out) {
  __shared__ float pT[JT * PT_STRIDE];        // p transposed: pT[j_local][i_local]
  __shared__ float whT[2][JT * WH_STRIDE];    // double-buffered Wh tile
  __shared__ float lsums[32 * 8];
  __shared__ float lsumtot[32];

  const int t       = threadIdx.x;
  const int wv      = t >> 5;
  const int lane    = t & 31;
  const int lhalf   = lane >> 4;
  const int lm      = lane & 15;
  const int rowhalf = (wv >> 2) & 1;
  const int nb      = (wv & 3) * 64;
  const int rb      = blockIdx.x * 32;

  // score-thread mapping: thread owns row i_loc, 4 consecutive j's
  const int il = t >> 3;                      // 0..31
  const int jg = t & 7;                       // 0..7 (x4 j each)
  const float s1i = s1[rb + il];
  const float xm  = s1i + smax[0];
  const float mrow = xm > 0.f ? xm : ALPHA * xm;   // exact row softmax max
  float lsum = 0.f;

  // per-thread piece of the tile DMA: 8 x 16B
  const int stg_r[1] = {0};
  v8f acc[4];
  #pragma unroll
  for (int i = 0; i < 4; ++i)
    acc[i] = (v8f){0.f,0.f,0.f,0.f,0.f,0.f,0.f,0.f};

  const size_t adj_row = (size_t)(rb + il) * NV;
  (void)stg_r;

  // ---- async staging of one 32x256 Wh tile into whT[buf] (8 insts/wave) ----
  auto stage_async = [&](int j0, int buf) {
    #pragma unroll
    for (int it = 0; it < 8; ++it) {
      const int idx = it * 1024 + t * 4;
      const int rj  = idx >> 8;
      const int cc  = idx & 255;
      const float* src = &Wh[(size_t)(j0 + rj) * F + cc];
      const unsigned ldsoff =
          (unsigned)(uintptr_t)&whT[buf][rj * WH_STRIDE + cc];
      asm volatile("global_load_async_to_lds_b128 %0, %1, off"
                   :: "v"(ldsoff), "v"(src) : "memory");
    }
  };

  stage_async(0, 0);                          // prologue: tile 0 -> buf 0

  for (int j0 = 0; j0 < NV; j0 += JT) {
    const int cur = (j0 >> 5) & 1;
    const bool has_next = (j0 + JT) < NV;
    if (has_next) stage_async(j0 + JT, cur ^ 1);   // overlap with tile j0 work

    // ---- masked scores -> p into LDS (A-fragment friendly, transposed) ----
    const int4   av  = *reinterpret_cast<const int4*>(&adj[adj_row + j0 + jg * 4]);
    const float4 s2v = *reinterpret_cast<const float4*>(&s2[j0 + jg * 4]);
    const float xs[4] = {s2v.x, s2v.y, s2v.z, s2v.w};
    const int   ad[4] = {av.x, av.y, av.z, av.w};
    #pragma unroll
    for (int q = 0; q < 4; ++q) {
      const float x = s1i + xs[q];
      const float e = x > 0.f ? x : ALPHA * x;
      const float p = (ad[q] > 0) ? expf(e - mrow) : 0.f;
      lsum += p;
      pT[(jg * 4 + q) * PT_STRIDE + il] = p;
    }
    if (has_next)  // gfx1250 global_prefetch_b8 of next adj tile
      __builtin_prefetch(&adj[adj_row + j0 + JT + jg * 4], 0, 0);

    // retire this wave's DMA for tile j0 (in-order completion: oldest 8 done),
    // keeping the next tile's 8 in flight; then make it WG-visible.
    if (has_next) asm volatile("s_wait_asynccnt 0x8" ::: "memory");
    else          asm volatile("s_wait_asynccnt 0x0" ::: "memory");
    __syncthreads();

    // ---- 32 WMMAs: acc[nt] += pT(16xK) @ whT(Kx16) over K=32 ----
    #pragma unroll
    for (int ks = 0; ks < 8; ++ks) {
      const int kr = ks * 4 + lhalf * 2;
      v2f afrag;
      afrag.x = pT[kr * PT_STRIDE + rowhalf * 16 + lm];
      afrag.y = pT[(kr + 1) * PT_STRIDE + rowhalf * 16 + lm];
      #pragma unroll
      for (int nt = 0; nt < 4; ++nt) {
        const int col = nb + nt * 16 + lm;
        v2f bfrag;
        bfrag.x = whT[cur][kr * WH_STRIDE + col];
        bfrag.y = whT[cur][(kr + 1) * WH_STRIDE + col];
        acc[nt] = __builtin_amdgcn_wmma_f32_16x16x4_f32(
            false, afrag, false, bfrag, (short)0, acc[nt], false, false);
      }
    }
    __syncthreads();   // done reading pT / whT[cur] before they are rewritten
  }

  // ---- reduce row sums ----
  lsums[il * 8 + jg] = lsum;
  __syncthreads();
  if (t < 32) {
    float s = 0.f;
    #pragma unroll
    for (int g = 0; g < 8; ++g) s += lsums[t * 8 + g];
    lsumtot[t] = s;
  }
  __syncthreads();

  // ---- normalize + ELU + store ----
  #pragma unroll
  for (int nt = 0; nt < 4; ++nt) {
    #pragma unroll
    for (int v = 0; v < 8; ++v) {
      const int Ml  = rowhalf * 16 + v + lhalf * 8;
      const float val = acc[nt][v] / lsumtot[Ml];
      const float o   = val > 0.f ? val : (expf(val) - 1.f);
      out[(size_t)(rb + Ml) * F + nb + nt * 16 + lm] = o;
    }
  }
}

// ---------------------------------------------------------------------------
extern "C" void kernel_launch(void* const* d_in, const int* in_sizes, int n_in,
                              void* d_out, int out_size, void* d_ws, size_t ws_size,
                              hipStream_t stream) {
  const float* h   = (const float*)d_in[0];
  const int*   adj = (const int*)d_in[1];
  const float* W   = (const float*)d_in[2];
  const float* a   = (const float*)d_in[3];
  float* out = (float*)d_out;

  float* ws   = (float*)d_ws;
  float* Wh   = ws;                          // 8192*256
  float* s1   = ws + (size_t)NV * F;         // 8192
  float* s2   = s1 + NV;                     // 8192
  float* smax = s2 + NV;                     // 1

  gat_gemm_hW<<<NV / 16, 128, 0, stream>>>(h, W, Wh);
  gat_scores<<<NV / 8, 256, 0, stream>>>(Wh, a, s1, s2);
  gat_s2max<<<1, 256, 0, stream>>>(s2, smax);
  gat_attn<<<NV / 32, 256, 0, stream>>>(Wh, adj, s1, s2, smax, out);
}